// ScopeWiseSum_82592221102251
// MI455X (gfx1250) — compile-verified
//
#include <hip/hip_runtime.h>

typedef __attribute__((ext_vector_type(2))) float v2f;
typedef __attribute__((ext_vector_type(8))) float v8f;

#define NUM_SCOPES 128
#define NUM_DECOMPS 8
#define BATCH 1024
#define K_IN 32
#define N_OUT 32

constexpr int WAVES   = 8;            // 256 threads / wave32
constexpr int STRIP   = 32;           // batch rows per strip (one row per lane)
constexpr int NSTRIPS = BATCH / STRIP;
constexpr int APAD    = 33;           // padded LDS row stride (floats) -> conflict-free

// One block per (scope, decomp). Computes out[sd] = logmatmul(x[sd], log_softmax weights).
__global__ __launch_bounds__(256)
void logmm_kernel(const float* __restrict__ x,
                  const float* __restrict__ acc,
                  float* __restrict__ out) {
  __shared__ float sB[K_IN * N_OUT];          // acc tile, then Bexp[k][j] = acc/colmax
  __shared__ float sMax[N_OUT];               // per-column max of acc
  __shared__ float sCb[N_OUT];                // max_b[j] = log(colmax) - log(colsum)
  __shared__ float sA[WAVES][STRIP * APAD];   // per-wave exp'd A strip (padded rows)

  const int tid  = threadIdx.x;
  const int lane = tid & 31;
  const int wave = tid >> 5;
  const int sd   = blockIdx.x;

  const float* xb = x   + (size_t)sd * BATCH * K_IN;
  const float* ab = acc + (size_t)sd * K_IN * N_OUT;
  float*       ob = out + (size_t)sd * BATCH * N_OUT;

  // ---- stage acc[s,d] (32x32 = 4KB) into LDS, coalesced float4 ----
  ((float4*)sB)[tid] = ((const float4*)ab)[tid];
  __syncthreads();

  // ---- per-column stats: colmax and colsum (bank-conflict-free column walk) ----
  if (tid < N_OUT) {
    const int j = tid;
    float m = sB[j], s = sB[j];
    #pragma unroll
    for (int k = 1; k < K_IN; ++k) {
      float v = sB[k * N_OUT + j];
      m = fmaxf(m, v);
      s += v;
    }
    sMax[j] = m;
    sCb[j]  = __logf(m) - __logf(s);   // log(colmax/colsum) == max_b[j]
  }
  __syncthreads();

  // ---- normalize in place: Bexp[k][j] = acc[k][j] / colmax[j]  (== exp(W - max_b)) ----
  {
    const int j = (tid * 4) & (N_OUT - 1);
    float4 v = ((float4*)sB)[tid];
    v.x /= sMax[j + 0]; v.y /= sMax[j + 1];
    v.z /= sMax[j + 2]; v.w /= sMax[j + 3];
    ((float4*)sB)[tid] = v;
  }
  __syncthreads();

  // ---- each wave caches all B fragments: b[jhalf][kstep], ISA 4x16 f32 layout ----
  // lane l: n = (l&15)+16*jh ; VGPR0 = B[k0][n], VGPR1 = B[k0+1][n], k0 = 4*kb + 2*(l>>4)
  const int ml    = lane & 15;
  const int khalf = lane >> 4;
  v2f bfrag[2][8];
  #pragma unroll
  for (int jh = 0; jh < 2; ++jh) {
    #pragma unroll
    for (int kb = 0; kb < 8; ++kb) {
      const int k0 = kb * 4 + 2 * khalf;
      const int n  = jh * 16 + ml;
      bfrag[jh][kb].x = sB[k0 * N_OUT + n];
      bfrag[jh][kb].y = sB[(k0 + 1) * N_OUT + n];
    }
  }
  const float cb0 = sCb[ml];
  const float cb1 = sCb[16 + ml];

  float* const aw = &sA[wave][0];

  for (int strip = wave; strip < NSTRIPS; strip += WAVES) {
    const int m0 = strip * STRIP;

    // ---- lane l owns batch row m0+l: load 32 f32 (8x b128), row max, exp ----
    const float4* rp = (const float4*)(xb + (size_t)(m0 + lane) * K_IN);
    float4 r[8];
    #pragma unroll
    for (int i = 0; i < 8; ++i) r[i] = rp[i];

    float amax = r[0].x;
    #pragma unroll
    for (int i = 0; i < 8; ++i)
      amax = fmaxf(amax, fmaxf(fmaxf(r[i].x, r[i].y), fmaxf(r[i].z, r[i].w)));

    float* arow = aw + lane * APAD;
    #pragma unroll
    for (int i = 0; i < 8; ++i) {
      arow[4 * i + 0] = __expf(r[i].x - amax);
      arow[4 * i + 1] = __expf(r[i].y - amax);
      arow[4 * i + 2] = __expf(r[i].z - amax);
      arow[4 * i + 3] = __expf(r[i].w - amax);
    }
    __builtin_amdgcn_wave_barrier();   // LDS is in-order per wave; just pin compiler order

    // ---- 4 independent WMMA accumulation chains: (mtile x jhalf), K = 32 in steps of 4 ----
    v8f c00 = {}, c01 = {}, c10 = {}, c11 = {};
    #pragma unroll
    for (int kb = 0; kb < 8; ++kb) {
      const int k0 = kb * 4 + 2 * khalf;
      v2f a0, a1;                                  // A frags for m-tiles 0 and 1
      a0.x = aw[ml        * APAD + k0];
      a0.y = aw[ml        * APAD + k0 + 1];
      a1.x = aw[(16 + ml) * APAD + k0];
      a1.y = aw[(16 + ml) * APAD + k0 + 1];
      c00 = __builtin_amdgcn_wmma_f32_16x16x4_f32(false, a0, false, bfrag[0][kb], (short)0, c00, false, false);
      c01 = __builtin_amdgcn_wmma_f32_16x16x4_f32(false, a0, false, bfrag[1][kb], (short)0, c01, false, false);
      c10 = __builtin_amdgcn_wmma_f32_16x16x4_f32(false, a1, false, bfrag[0][kb], (short)0, c10, false, false);
      c11 = __builtin_amdgcn_wmma_f32_16x16x4_f32(false, a1, false, bfrag[1][kb], (short)0, c11, false, false);
    }

    // ---- epilogue on C layout: out = log(prod) + max_a[m] + max_b[j]; coalesced stores ----
    #pragma unroll
    for (int mt = 0; mt < 2; ++mt) {
      const v8f cA = mt ? c10 : c00;   // j = 0..15
      const v8f cB = mt ? c11 : c01;   // j = 16..31
      #pragma unroll
      for (int v = 0; v < 8; ++v) {
        const int mlocal = mt * 16 + v + 8 * khalf;       // C-layout row for this lane/VGPR
        const float am   = __shfl(amax, mlocal, 32);      // owner lane of that row's max
        float* orow = ob + (size_t)(m0 + mlocal) * N_OUT;
        orow[ml]      = __logf(cA[v]) + am + cb0;
        orow[16 + ml] = __logf(cB[v]) + am + cb1;
      }
    }
    __builtin_amdgcn_wave_barrier();   // next iteration reuses aw (same-wave, in-order LDS)
  }
}

extern "C" void kernel_launch(void* const* d_in, const int* in_sizes, int n_in,
                              void* d_out, int out_size, void* d_ws, size_t ws_size,
                              hipStream_t stream) {
  const float* x   = (const float*)d_in[0];
  const float* acc = (const float*)d_in[1];
  float* out       = (float*)d_out;

  dim3 grid(NUM_SCOPES * NUM_DECOMPS);   // 1024 blocks, one per (scope, decomp)
  dim3 block(256);                       // 8 wave32
  hipLaunchKernelGGL(logmm_kernel, grid, block, 0, stream, x, acc, out);
}